// GCN_452
// MI455X (gfx1250) — compile-verified
//
#include <hip/hip_runtime.h>
#include <hip/hip_bf16.h>

typedef __attribute__((ext_vector_type(2))) float v2f;
typedef __attribute__((ext_vector_type(8))) float v8f;
typedef __attribute__((ext_vector_type(4))) unsigned int u32x4;
typedef __attribute__((ext_vector_type(8))) int i32x8;
typedef __attribute__((ext_vector_type(4))) int i32x4;

// Non-returning FP32 atomic add -> global_atomic_add_f32 (STOREcnt path).
__device__ __forceinline__ void fatomic_add(float* p, float v) {
    __hip_atomic_fetch_add(p, v, __ATOMIC_RELAXED, __HIP_MEMORY_SCOPE_AGENT);
}

__global__ void zero_f(float* __restrict__ p, int n) {
    int i = blockIdx.x * blockDim.x + threadIdx.x;
    if (i < n) p[i] = 0.0f;
}

__global__ void count_deg(const int* __restrict__ dst, float* __restrict__ deg, int E) {
    int i = blockIdx.x * blockDim.x + threadIdx.x;
    if (i < E) fatomic_add(&deg[dst[i]], 1.0f);
}

// deg includes implicit self-loop: dinv = rsqrt(edge_deg + 1)
__global__ void finalize_dinv(float* __restrict__ d, int n) {
    int i = blockIdx.x * blockDim.x + threadIdx.x;
    if (i < n) d[i] = rsqrtf(d[i] + 1.0f);
}

__global__ void relu_ip(float* __restrict__ p, long long n) {
    long long i = (long long)blockIdx.x * blockDim.x + threadIdx.x;
    if (i < n) p[i] = fmaxf(p[i], 0.0f);
}

// H[N x CPAD] = A[N x K] @ W[K x COUT]  (CPAD = NT*16; pad cols get zeros).
// blockDim = (32, NT): wave `wv` owns output col tile wv; each block does RT row tiles.
// W is staged K x COUT into LDS via the Tensor Data Mover (one D# per block),
// then the K-loop reads branch-free B fragments from LDS:
//   FP32 WMMA V_WMMA_F32_16X16X4_F32, K-step 4 (ISA 7.12.2 layouts):
//   A 16x4: lanes 0-15 hold K=0,1 in v0,v1; lanes 16-31 hold K=2,3. B 4x16 mirrored.
//   C/D: VGPR r = row r (lanes 0-15) / row r+8 (lanes 16-31).
template <int K, int COUT, int NT, int RT>
__global__ __launch_bounds__(32 * NT)
void gemm_wmma(const float* __restrict__ A, const float* __restrict__ W,
               float* __restrict__ H, int nrows) {
    constexpr int CPAD = NT * 16;
    __shared__ float sW[K * COUT];

    const int lane  = threadIdx.x;     // 0..31
    const int wv    = threadIdx.y;     // output col tile
    const int r16   = lane & 15;
    const int hi    = lane >> 4;       // 0/1
    const int khalf = hi * 2;          // 0 or 2

    // ---- stage W (K x COUT f32, row-major) into LDS ----
#if __has_builtin(__builtin_amdgcn_tensor_load_to_lds) && __has_builtin(__builtin_amdgcn_s_wait_tensorcnt)
    if (wv == 0) {
        unsigned lds_base =
            (unsigned)(unsigned long long)(__attribute__((address_space(3))) float*)sW;
        unsigned long long ga = (unsigned long long)(const void*)W;
        // D# group 0: count=1 | lds_addr | global_addr[56:0] | type=2
        u32x4 g0;
        g0.x = 1u;
        g0.y = lds_base;
        g0.z = (unsigned)ga;
        g0.w = (unsigned)((ga >> 32) & 0x01FFFFFFu) | (2u << 30);
        // D# group 1: data_size=4B; tensor COUT x K; tile COUT x K; dim0 stride COUT
        i32x8 g1;
        g1[0] = (int)(2u << 16);                         // data_size = 2 (4 bytes)
        g1[1] = (int)(((unsigned)COUT & 0xFFFFu) << 16); // tensor_dim0[15:0]
        g1[2] = (int)(((unsigned)K & 0xFFFFu) << 16);    // tensor_dim0[31:16]=0 | tensor_dim1[15:0]
        g1[3] = (int)(((unsigned)COUT & 0xFFFFu) << 16); // tensor_dim1[31:16]=0 | tile_dim0
        g1[4] = K;                                       // tile_dim1 | tile_dim2=0 (2D)
        g1[5] = COUT;                                    // tensor_dim0_stride[31:0]
        g1[6] = 0;                                       // stride hi | dim1_stride lo (2D: unused)
        g1[7] = 0;
        i32x4 gz4 = {0, 0, 0, 0};                        // groups 2/3 unused (2D tensor)
        i32x8 gz8 = {0, 0, 0, 0, 0, 0, 0, 0};            // extra group (6-arg toolchain), unused
        __builtin_amdgcn_tensor_load_to_lds(g0, g1, gz4, gz4, gz8, 0);
        __builtin_amdgcn_s_wait_tensorcnt(0);
    }
    __syncthreads();
#else
    for (int i = lane + 32 * wv; i < K * COUT; i += 32 * NT) sW[i] = W[i];
    __syncthreads();
#endif

    // ---- per-lane A pointers for RT row tiles (clamped: EXEC stays all-ones) ----
    const int rowbase = blockIdx.x * (RT * 16);
    const float* ap[RT];
#pragma unroll
    for (int t = 0; t < RT; ++t) {
        int r = rowbase + t * 16 + r16;
        if (r >= nrows) r = nrows - 1;
        ap[t] = A + (size_t)r * K + khalf;
    }

    const int col  = wv * 16 + r16;
    const bool cok = (col < COUT);                 // compile-time true unless padded tile
    const int colc = cok ? col : (COUT - 1);

    v8f acc[RT] = {};
#pragma unroll 4
    for (int k0 = 0; k0 < K; k0 += 4) {
        const int k = k0 + khalf;
        v2f b;
        float b0 = sW[k * COUT + colc];
        float b1 = sW[(k + 1) * COUT + colc];
        b.x = cok ? b0 : 0.0f;                     // v_cndmask, no branch
        b.y = cok ? b1 : 0.0f;
#pragma unroll
        for (int t = 0; t < RT; ++t) {
            v2f a;
            a.x = ap[t][k0];
            a.y = ap[t][k0 + 1];
            acc[t] = __builtin_amdgcn_wmma_f32_16x16x4_f32(
                false, a, false, b, (short)0, acc[t], false, false);
        }
    }

    // ---- epilogue: uniform per-tile branch; full tiles store unconditionally ----
#pragma unroll
    for (int t = 0; t < RT; ++t) {
        const int rtile = rowbase + t * 16;
        const int mb    = rtile + hi * 8;
        float* hp = H + (size_t)mb * CPAD + wv * 16 + r16;
        if (rtile + 16 <= nrows) {                 // uniform (SALU) branch: fast path
#pragma unroll
            for (int r = 0; r < 8; ++r) hp[(size_t)r * CPAD] = acc[t][r];
        } else {                                   // tail tile of the last block only
#pragma unroll
            for (int r = 0; r < 8; ++r) {
                if (mb + r < nrows) hp[(size_t)r * CPAD] = acc[t][r];
            }
        }
    }
}

// agg[v][c] = bias[c] + dinv[v]^2 * h[v][c]   (self-loop term + bias fused)
template <int C, int HP>
__global__ void init_agg(const float* __restrict__ h, const float* __restrict__ dinv,
                         const float* __restrict__ bias, float* __restrict__ agg, int n) {
    long long idx = (long long)blockIdx.x * blockDim.x + threadIdx.x;
    if (idx >= (long long)n * C) return;
    int node = (int)(idx / C);
    int c    = (int)(idx - (long long)node * C);
    float di = dinv[node];
    agg[idx] = bias[c] + di * di * h[(size_t)node * HP + c];
}

// agg[dst][c] += dinv[src]*dinv[dst] * h[src][c]; one thread per (edge, 4 channels)
template <int C, int HP>
__global__ void edge_scatter(const int* __restrict__ src, const int* __restrict__ dst,
                             const float* __restrict__ dinv, const float* __restrict__ h,
                             float* __restrict__ agg, int E) {
    constexpr int G = C / 4;
    long long gid = (long long)blockIdx.x * blockDim.x + threadIdx.x;
    long long e = gid / G;
    int g = (int)(gid - e * G);
    if (e >= E) return;
    const int s = src[e];
    const int d = dst[e];
    const float norm = dinv[s] * dinv[d];
    const float4 hv = *(const float4*)(h + (size_t)s * HP + g * 4);
    float* out = agg + (size_t)d * C + g * 4;
    fatomic_add(out + 0, norm * hv.x);
    fatomic_add(out + 1, norm * hv.y);
    fatomic_add(out + 2, norm * hv.z);
    fatomic_add(out + 3, norm * hv.w);
}

static inline unsigned int cdiv_u(long long a, int b) {
    return (unsigned int)((a + b - 1) / b);
}

extern "C" void kernel_launch(void* const* d_in, const int* in_sizes, int n_in,
                              void* d_out, int out_size, void* d_ws, size_t ws_size,
                              hipStream_t stream) {
    constexpr int IN_C = 128, HID_C = 64, OUT_C = 40, OUT_P = 48;
    constexpr int RT = 4;  // row tiles per block

    const float* x  = (const float*)d_in[0];
    const int*   ei = (const int*)d_in[1];
    const float* W1 = (const float*)d_in[2];
    const float* b1 = (const float*)d_in[3];
    const float* W2 = (const float*)d_in[4];
    const float* b2 = (const float*)d_in[5];
    const float* W3 = (const float*)d_in[6];
    const float* b3 = (const float*)d_in[7];
    float* out = (float*)d_out;

    const int N = in_sizes[0] / IN_C;
    const int E = in_sizes[1] / 2;
    const int* src = ei;       // edge_index[0]
    const int* dst = ei + E;   // edge_index[1]

    // Workspace: dinv[N] | h[N*64] | agg[N*64]  (~52 MB)
    char* ws = (char*)d_ws;
    float* dinv = (float*)ws;
    size_t off = (((size_t)N * 4) + 255) & ~(size_t)255;
    float* h = (float*)(ws + off);
    off += (size_t)N * HID_C * 4;
    float* agg = (float*)(ws + off);

    const int B = 256;
    const unsigned int gemmBlocks = cdiv_u(N, 16 * RT);

    // --- normalization: dinv = rsqrt(deg + 1) ---
    zero_f<<<cdiv_u(N, B), B, 0, stream>>>(dinv, N);
    count_deg<<<cdiv_u(E, B), B, 0, stream>>>(dst, dinv, E);
    finalize_dinv<<<cdiv_u(N, B), B, 0, stream>>>(dinv, N);

    // --- layer 1: x @ W1, aggregate, relu ---
    gemm_wmma<IN_C, HID_C, 4, RT><<<gemmBlocks, dim3(32, 4), 0, stream>>>(x, W1, h, N);
    init_agg<HID_C, HID_C><<<cdiv_u((long long)N * HID_C, B), B, 0, stream>>>(h, dinv, b1, agg, N);
    edge_scatter<HID_C, HID_C><<<cdiv_u((long long)E * (HID_C / 4), B), B, 0, stream>>>(
        src, dst, dinv, h, agg, E);
    relu_ip<<<cdiv_u((long long)N * HID_C, B), B, 0, stream>>>(agg, (long long)N * HID_C);

    // --- layer 2: agg @ W2, aggregate, relu ---
    gemm_wmma<HID_C, HID_C, 4, RT><<<gemmBlocks, dim3(32, 4), 0, stream>>>(agg, W2, h, N);
    init_agg<HID_C, HID_C><<<cdiv_u((long long)N * HID_C, B), B, 0, stream>>>(h, dinv, b2, agg, N);
    edge_scatter<HID_C, HID_C><<<cdiv_u((long long)E * (HID_C / 4), B), B, 0, stream>>>(
        src, dst, dinv, h, agg, E);
    relu_ip<<<cdiv_u((long long)N * HID_C, B), B, 0, stream>>>(agg, (long long)N * HID_C);

    // --- layer 3: agg @ W3 (40 cols padded to 48), aggregate straight into d_out ---
    gemm_wmma<HID_C, OUT_C, 3, RT><<<gemmBlocks, dim3(32, 3), 0, stream>>>(agg, W3, h, N);
    init_agg<OUT_C, OUT_P><<<cdiv_u((long long)N * OUT_C, B), B, 0, stream>>>(h, dinv, b3, out, N);
    edge_scatter<OUT_C, OUT_P><<<cdiv_u((long long)E * (OUT_C / 4), B), B, 0, stream>>>(
        src, dst, dinv, h, out, E);
}